// MyBCELossLogits_79465484910812
// MI455X (gfx1250) — compile-verified
//
#include <hip/hip_runtime.h>

typedef float v2f __attribute__((ext_vector_type(2)));
typedef float v4f __attribute__((ext_vector_type(4)));
typedef float v8f __attribute__((ext_vector_type(8)));
typedef int   v4i __attribute__((ext_vector_type(4)));

#define SCALE_W 8.0f

// ---------------------------------------------------------------------------
// Wave32 full-lane sum using V_WMMA_F32_16X16X4_F32.
// A (16x4 f32): VGPR0 holds K=0 for lanes 0-15 and K=2 for lanes 16-31,
// VGPR1 (K=1/K=3) zeroed. With B = all-ones (4x16):
//   D[m][n] = acc[lane m] + acc[lane m+16]  for every column n.
// Each lane sums its 8 C VGPRs (lanes 0-15 cover rows 0-7, lanes 16-31 rows
// 8-15); one xor-16 shuffle then yields the full 32-lane sum on every lane.
// REQUIRES EXEC == all ones (call only from wave-uniform control flow).
// ---------------------------------------------------------------------------
__device__ __forceinline__ float wave_sum32(float v) {
    v2f a; a[0] = v;    a[1] = 0.0f;
    v2f b; b[0] = 1.0f; b[1] = 1.0f;
    v8f c = {};
    c = __builtin_amdgcn_wmma_f32_16x16x4_f32(
            /*neg_a=*/false, a, /*neg_b=*/false, b,
            /*c_mod=*/(short)0, c, /*reuse_a=*/false, /*reuse_b=*/false);
    float s = ((c[0] + c[1]) + (c[2] + c[3])) + ((c[4] + c[5]) + (c[6] + c[7]));
    s += __shfl_xor(s, 16, 32);
    return s;
}

// ---------------------------------------------------------------------------
// Branchless weighted BCE-with-logits element:
//   L        = log1p(exp(-|x|))                  (shared exp+log)
//   softplus(-x) = L + 0.5(|x|-x);  softplus(x) = L + 0.5(|x|+x)
//   relu     = select on t  (one cndmask, picks which softplus we build)
//   weight   = 1 if sign-prediction matches target else SCALE (one cndmask)
//   guard    = 0 unless t in {0,1}               (one cndmask)
// Everything is straight-line VALU: no exec-mask divergence in the hot loop.
// ---------------------------------------------------------------------------
__device__ __forceinline__ float elem_loss(float x, int t) {
    float ax      = __builtin_fabsf(x);
    float e       = __expf(-ax);              // v_mul + v_exp_f32
    float L       = __logf(1.0f + e);         // v_add + v_log_f32 (+mul), 1+e in [1,2]
    float half_ax = 0.5f * ax;
    float half_x  = 0.5f * x;
    bool  tpos    = (t == 1);
    float relu    = tpos ? (half_ax - half_x) : (half_ax + half_x); // cndmask
    bool  hit     = ((x >= 0.0f) == tpos);    // prediction matches target
    float w       = hit ? 1.0f : SCALE_W;     // cndmask
    float r       = (L + relu) * w;
    return ((unsigned)t <= 1u) ? r : 0.0f;    // cndmask guard (t outside {0,1})
}

__device__ __forceinline__ float granule4(const v4f* __restrict__ x4,
                                          const v4i* __restrict__ t4, int i) {
    v4f x = __builtin_nontemporal_load(&x4[i]);   // global_load_b128, TH=NT
    v4i t = __builtin_nontemporal_load(&t4[i]);   // global_load_b128, TH=NT
    return (elem_loss(x[0], t[0]) + elem_loss(x[1], t[1])) +
           (elem_loss(x[2], t[2]) + elem_loss(x[3], t[3]));
}

// ---------------------------------------------------------------------------
// Stage 1: grid-stride streaming, 2x unrolled with dual accumulators
// (4 independent b128 NT loads in flight per lane per iteration).
// Per-lane accumulate -> WMMA wave reduce -> LDS combine of 8 waves ->
// one partial per block. Bandwidth-bound: 512 MiB @ 23.3 TB/s ~ 23 us.
// ---------------------------------------------------------------------------
__global__ void __launch_bounds__(256)
bce_partial_kernel(const float* __restrict__ logits,
                   const int*   __restrict__ targets,
                   float*       __restrict__ partial,
                   int n4) {
    __shared__ float smem[8];

    const v4f* __restrict__ x4 = (const v4f*)logits;
    const v4i* __restrict__ t4 = (const v4i*)targets;

    const int stride  = (int)(gridDim.x * blockDim.x);
    const int stride2 = stride * 2;

    float acc0 = 0.0f, acc1 = 0.0f;
    int i = (int)(blockIdx.x * blockDim.x + threadIdx.x);
    for (; i + stride < n4; i += stride2) {
        acc0 += granule4(x4, t4, i);
        acc1 += granule4(x4, t4, i + stride);
    }
    if (i < n4) acc0 += granule4(x4, t4, i);

    // wave-level reduce (EXEC all ones: no divergence at this point)
    float ws = wave_sum32(acc0 + acc1);

    const int lane = (int)(threadIdx.x & 31u);
    const int wave = (int)(threadIdx.x >> 5u);
    if (lane == 0) smem[wave] = ws;
    __syncthreads();

    if (wave == 0) {                       // wave-uniform branch: EXEC all ones
        float v  = (lane < 8) ? smem[lane] : 0.0f;
        float bs = wave_sum32(v);
        if (lane == 0) partial[blockIdx.x] = bs;
    }
}

// ---------------------------------------------------------------------------
// Stage 2: one block reduces the block partials and applies the 1/N mean.
// ---------------------------------------------------------------------------
__global__ void __launch_bounds__(256)
bce_final_kernel(const float* __restrict__ partial,
                 int nblocks,
                 float* __restrict__ out,
                 float inv_n) {
    __shared__ float smem[8];

    float acc = 0.0f;
    for (int i = (int)threadIdx.x; i < nblocks; i += (int)blockDim.x)
        acc += partial[i];

    float ws = wave_sum32(acc);

    const int lane = (int)(threadIdx.x & 31u);
    const int wave = (int)(threadIdx.x >> 5u);
    if (lane == 0) smem[wave] = ws;
    __syncthreads();

    if (wave == 0) {
        float v  = (lane < 8) ? smem[lane] : 0.0f;
        float bs = wave_sum32(v);
        if (lane == 0) out[0] = bs * inv_n;
    }
}

extern "C" void kernel_launch(void* const* d_in, const int* in_sizes, int n_in,
                              void* d_out, int out_size, void* d_ws, size_t ws_size,
                              hipStream_t stream) {
    const float* logits  = (const float*)d_in[0];
    const int*   targets = (const int*)d_in[1];
    float*       out     = (float*)d_out;
    float*       partial = (float*)d_ws;

    const int n  = in_sizes[0];      // 2^26
    const int n4 = n >> 2;           // float4 / int4 granules

    const int BLOCK   = 256;         // 8 wave32 waves per block
    const int NBLOCKS = 4096;        // ~1M lanes -> 16 KiB of partials in d_ws

    bce_partial_kernel<<<NBLOCKS, BLOCK, 0, stream>>>(logits, targets, partial, n4);
    bce_final_kernel<<<1, BLOCK, 0, stream>>>(partial, NBLOCKS, out, 1.0f / (float)n);
}